// IC_18004502905384
// MI455X (gfx1250) — compile-verified
//
#include <hip/hip_runtime.h>
#include <hip/hip_bf16.h>
#include <math.h>

// Problem constants (from reference): N=100000 nodes, E=16000000 edges, 10 steps.
#define IC_STEPS 10

typedef int   v4i __attribute__((ext_vector_type(4)));
typedef float v4f __attribute__((ext_vector_type(4)));

// ---------------------------------------------------------------------------
// PROBE (never launched): placed FIRST so the disasm snippet shows exactly how
// the three performance-critical constructs lower on gfx1250:
//   - __builtin_nontemporal_load (expect global_load_b128 ... th:TH_LOAD_NT)
//   - __builtin_prefetch         (expect global_prefetch_b8)
//   - __hip_atomic_fetch_add f32 (expect global_atomic_add_f32, NOT a CAS loop)
// Uses only threadIdx.x so the prologue is minimal and the body fits the
// snippet window.
// ---------------------------------------------------------------------------
__global__ void aa_probe(const int* __restrict__ idx,
                         const float* __restrict__ val,
                         float* __restrict__ agg) {
  int i = threadIdx.x;
  const v4i* p = (const v4i*)idx + i;
  __builtin_prefetch(p + 64, 0, 0);
  v4i r4 = __builtin_nontemporal_load(p);
  float v = val[r4[0]];
  __hip_atomic_fetch_add(&agg[r4[1]], v,
                         __ATOMIC_RELAXED, __HIP_MEMORY_SCOPE_AGENT);
}

// ---------------------------------------------------------------------------
// Edge pass (int32 indices): 4 edges/thread/iter via 128-bit NT loads.
//   t   = w * x[row]            (gather hits L2: x is 400 KB)
//   msg = log1p(EPS - t)
//   agg[col] += msg             (fire-and-forget global_atomic_add_f32)
// Edge streams are read-once-per-step -> non-temporal so they don't evict the
// hot node state from L2. Prefetch the next grid-stride iteration.
// ---------------------------------------------------------------------------
__global__ void ic_edge_i32(const int* __restrict__ rows,
                            const int* __restrict__ cols,
                            const float* __restrict__ w,
                            const float* __restrict__ x,
                            float* __restrict__ agg, int nvec) {
  int i = blockIdx.x * blockDim.x + threadIdx.x;
  int stride = gridDim.x * blockDim.x;
  const v4i* __restrict__ r4p = (const v4i*)rows;
  const v4i* __restrict__ c4p = (const v4i*)cols;
  const v4f* __restrict__ w4p = (const v4f*)w;
  for (; i < nvec; i += stride) {
    int pf = i + stride;
    if (pf < nvec) {                       // global_prefetch_b8 (gfx1250)
      __builtin_prefetch(r4p + pf, 0, 0);
      __builtin_prefetch(c4p + pf, 0, 0);
      __builtin_prefetch(w4p + pf, 0, 0);
    }
    v4i r4 = __builtin_nontemporal_load(r4p + i);   // global_load_b128 th:NT
    v4i c4 = __builtin_nontemporal_load(c4p + i);
    v4f w4 = __builtin_nontemporal_load(w4p + i);
#pragma unroll
    for (int k = 0; k < 4; ++k) {
      float t = w4[k] * x[r4[k]];
      float m = log1pf(1e-15f - t);
      // relaxed / agent scope, result unused -> no-return global_atomic_add_f32
      __hip_atomic_fetch_add(&agg[c4[k]], m,
                             __ATOMIC_RELAXED, __HIP_MEMORY_SCOPE_AGENT);
    }
  }
}

// ---------------------------------------------------------------------------
// Init: s = 1 - x0, x = x0, r = 0, agg = 0.  State lives directly in d_out.
// ---------------------------------------------------------------------------
__global__ void ic_init(const float* __restrict__ x0,
                        float* __restrict__ s, float* __restrict__ x,
                        float* __restrict__ r, float* __restrict__ agg, int n) {
  int i = blockIdx.x * blockDim.x + threadIdx.x;
  if (i < n) {
    float v = x0[i];
    s[i]   = 1.0f - v;
    x[i]   = v;
    r[i]   = 0.0f;
    agg[i] = 0.0f;
  }
}

// ---------------------------------------------------------------------------
// One-time index compaction: int64 -> int32 (covers rows then cols, 2E elems).
// Halves the steady-state index traffic (256 MB -> 128 MB per step).
// ---------------------------------------------------------------------------
__global__ void ic_cvt_idx(const long long* __restrict__ in64,
                           int* __restrict__ out32, long long n) {
  long long i = (long long)blockIdx.x * blockDim.x + threadIdx.x;
  long long stride = (long long)gridDim.x * blockDim.x;
  for (; i < n; i += stride) out32[i] = (int)in64[i];
}

// Fallback edge pass reading int64 indices directly (used only if d_ws cannot
// hold the compacted indices or E is not a multiple of 4; still needs N*4 bytes of d_ws for agg).
__global__ void ic_edge_i64(const long long* __restrict__ rows,
                            const long long* __restrict__ cols,
                            const float* __restrict__ w,
                            const float* __restrict__ x,
                            float* __restrict__ agg, long long ne) {
  long long i = (long long)blockIdx.x * blockDim.x + threadIdx.x;
  long long stride = (long long)gridDim.x * blockDim.x;
  for (; i < ne; i += stride) {
    int ri = (int)rows[i];
    int ci = (int)cols[i];
    float t = w[i] * x[ri];
    float m = log1pf(1e-15f - t);
    __hip_atomic_fetch_add(&agg[ci], m,
                           __ATOMIC_RELAXED, __HIP_MEMORY_SCOPE_AGENT);
  }
}

// ---------------------------------------------------------------------------
// Node pass: q = exp(agg); (s,x,r) <- (s*q, s*(1-q), r+x); agg <- 0 for the
// next step (saves a separate memset pass over the node buffer).
// ---------------------------------------------------------------------------
__global__ void ic_node(float* __restrict__ s, float* __restrict__ x,
                        float* __restrict__ r, float* __restrict__ agg, int n) {
  int i = blockIdx.x * blockDim.x + threadIdx.x;
  if (i < n) {
    float q = expf(agg[i]);
    agg[i] = 0.0f;
    float sv = s[i];
    float xv = x[i];
    r[i] += xv;
    s[i] = sv * q;
    x[i] = sv * (1.0f - q);
  }
}

extern "C" void kernel_launch(void* const* d_in, const int* in_sizes, int n_in,
                              void* d_out, int out_size, void* d_ws, size_t ws_size,
                              hipStream_t stream) {
  // Inputs per reference: edge_index (2,E) int64, edge_weight (E,1) f32, x0 (N,1) f32
  const long long* edge_index  = (const long long*)d_in[0];
  const float*     edge_weight = (const float*)d_in[1];
  const float*     x0          = (const float*)d_in[2];

  const long long E = (long long)in_sizes[0] / 2;   // 16M
  const int       n = in_sizes[2];                  // 100000

  // Output layout: (s | x | r), each n floats.
  float* s = (float*)d_out;
  float* x = s + n;
  float* r = s + 2 * n;

  // Workspace layout (preferred path): rows32[E] | cols32[E] | agg[n]
  const size_t need32 = (size_t)2 * (size_t)E * sizeof(int) + (size_t)n * sizeof(float);
  const bool use32 = (ws_size >= need32) && ((E & 3LL) == 0LL);

  int*   rows32 = nullptr;
  int*   cols32 = nullptr;
  float* agg    = nullptr;
  if (use32) {
    rows32 = (int*)d_ws;
    cols32 = rows32 + E;
    agg    = (float*)(cols32 + E);
  } else {
    agg = (float*)d_ws;   // needs only n*4 bytes
  }

  const int tb   = 256;                 // 8 wave32 waves per workgroup
  const int nblk = (n + tb - 1) / tb;

  ic_init<<<nblk, tb, 0, stream>>>(x0, s, x, r, agg, n);

  if (use32) {
    // One-time int64 -> int32 compaction of both rows and cols (2E elements).
    ic_cvt_idx<<<4096, tb, 0, stream>>>(edge_index, rows32, 2LL * E);

    const int nvec = (int)(E >> 2);     // 4M vec4 packets
    const int eblk = 8192;              // 2M threads -> 2 grid-stride iters
    for (int step = 0; step < IC_STEPS; ++step) {
      ic_edge_i32<<<eblk, tb, 0, stream>>>(rows32, cols32, edge_weight, x, agg, nvec);
      ic_node<<<nblk, tb, 0, stream>>>(s, x, r, agg, n);
    }
  } else {
    const long long* rows = edge_index;
    const long long* cols = edge_index + E;
    const int eblk = 8192;
    for (int step = 0; step < IC_STEPS; ++step) {
      ic_edge_i64<<<eblk, tb, 0, stream>>>(rows, cols, edge_weight, x, agg, E);
      ic_node<<<nblk, tb, 0, stream>>>(s, x, r, agg, n);
    }
  }
}